// GCN_91319594648115
// MI455X (gfx1250) — compile-verified
//
#include <hip/hip_runtime.h>

#define N_NODES      100000
#define N_EDGES      3200000
#define NUM_FEATURES 512
#define HIDDEN       16
#define NUM_CLASSES  32
#define E_TOT        (N_EDGES + N_NODES)   // edges + self loops

typedef __attribute__((ext_vector_type(2))) float v2f;
typedef __attribute__((ext_vector_type(8))) float v8f;

// ---------------------------------------------------------------------------
// Init: deg = 0, agg1 = broadcast(b1), out = broadcast(b2)
// ---------------------------------------------------------------------------
__global__ void gcn_init_kernel(float* __restrict__ deg,
                                float* __restrict__ agg1,
                                float* __restrict__ out,
                                const float* __restrict__ b1,
                                const float* __restrict__ b2) {
    int i = blockIdx.x * blockDim.x + threadIdx.x;
    if (i < N_NODES)              deg[i]  = 0.0f;
    if (i < N_NODES * HIDDEN)     agg1[i] = b1[i & (HIDDEN - 1)];
    if (i < N_NODES * NUM_CLASSES) out[i] = b2[i & (NUM_CLASSES - 1)];
}

// ---------------------------------------------------------------------------
// Degree: count dst occurrences over edges + self-loops (as float)
// ---------------------------------------------------------------------------
__global__ void gcn_degree_kernel(const long long* __restrict__ ei,
                                  float* __restrict__ deg) {
    int t = blockIdx.x * blockDim.x + threadIdx.x;
    if (t >= E_TOT) return;
    int d = (t < N_EDGES) ? (int)ei[N_EDGES + t] : (t - N_EDGES);
    atomicAdd(&deg[d], 1.0f);
}

// deg -> D^{-1/2} in place
__global__ void gcn_dinv_kernel(float* __restrict__ deg) {
    int i = blockIdx.x * blockDim.x + threadIdx.x;
    if (i >= N_NODES) return;
    float d = deg[i];
    deg[i] = (d > 0.0f) ? __frsqrt_rn(d) : 0.0f;
}

// ---------------------------------------------------------------------------
// GEMM1: xw[N,16] = x[N,512] @ W1[512,16] via V_WMMA_F32_16X16X4_F32
// One wave per 16-row tile (N_NODES = 6250 * 16 exactly). W1 staged in LDS.
// A layout (32-bit 16x4): lanes 0-15 hold K=k0,k0+1 of row M=lane;
//                         lanes 16-31 hold K=k0+2,k0+3 of row M=lane-16.
// C/D layout: VGPR r: lanes 0-15 -> M=r, N=lane; lanes 16-31 -> M=r+8.
// ---------------------------------------------------------------------------
__global__ void gcn_gemm1_wmma(const float* __restrict__ x,
                               const float* __restrict__ W1,
                               float* __restrict__ xw) {
    __shared__ float sW[NUM_FEATURES * HIDDEN];   // 32 KB of the 320 KB WGP LDS
    const int tid = threadIdx.x;
    for (int i = tid; i < NUM_FEATURES * HIDDEN; i += blockDim.x)
        sW[i] = W1[i];
    __syncthreads();

    const int wave = blockIdx.x * (blockDim.x >> 5) + (tid >> 5);
    const int lane = tid & 31;
    if (wave * 16 >= N_NODES) return;             // uniform per wave

    const int half = lane >> 4;                   // 0: K lo pair, 1: K hi pair
    const int m    = lane & 15;
    const float* xrow = x + (size_t)(wave * 16 + m) * NUM_FEATURES + 2 * half;

    v8f c = {0.f, 0.f, 0.f, 0.f, 0.f, 0.f, 0.f, 0.f};
#pragma unroll 8
    for (int k0 = 0; k0 < NUM_FEATURES; k0 += 4) {
        v2f a = *(const v2f*)(xrow + k0);         // 8B coalesced load
        const int kr = k0 + 2 * half;
        v2f b;
        b.x = sW[kr * HIDDEN + m];
        b.y = sW[(kr + 1) * HIDDEN + m];
        c = __builtin_amdgcn_wmma_f32_16x16x4_f32(
                false, a, false, b, (short)0, c, false, false);
    }

#pragma unroll
    for (int r = 0; r < 8; ++r) {
        const int rowD = wave * 16 + r + 8 * half;
        xw[(size_t)rowD * HIDDEN + m] = c[r];
    }
}

// ---------------------------------------------------------------------------
// Scatter layer 1: agg1[dst] += xw[src] * dinv[src]*dinv[dst]  (16 features)
// ---------------------------------------------------------------------------
__global__ void gcn_scatter1_kernel(const long long* __restrict__ ei,
                                    const float* __restrict__ dinv,
                                    const float* __restrict__ xw,
                                    float* __restrict__ agg1) {
    int t = blockIdx.x * blockDim.x + threadIdx.x;
    if (t >= E_TOT) return;
    int s, d;
    if (t < N_EDGES) { s = (int)ei[t]; d = (int)ei[N_EDGES + t]; }
    else             { s = t - N_EDGES; d = s; }
    const float nrm = dinv[s] * dinv[d];
    const float4* xs = (const float4*)(xw + (size_t)s * HIDDEN);
    float* ad = agg1 + (size_t)d * HIDDEN;
#pragma unroll
    for (int q = 0; q < 4; ++q) {
        float4 v = xs[q];                          // 16B coalesced gather
        atomicAdd(ad + 4 * q + 0, v.x * nrm);
        atomicAdd(ad + 4 * q + 1, v.y * nrm);
        atomicAdd(ad + 4 * q + 2, v.z * nrm);
        atomicAdd(ad + 4 * q + 3, v.w * nrm);
    }
}

// h = relu(agg1) (bias already folded into agg1 init); h lives in d_out tail
__global__ void gcn_relu_kernel(const float* __restrict__ agg1,
                                float* __restrict__ h) {
    int i = blockIdx.x * blockDim.x + threadIdx.x;
    if (i >= N_NODES * HIDDEN) return;
    float v = agg1[i];
    h[i] = v > 0.0f ? v : 0.0f;
}

// ---------------------------------------------------------------------------
// GEMM2: hw[N,32] = h[N,16] @ W2[16,32]; two 16x16 accumulators per wave
// ---------------------------------------------------------------------------
__global__ void gcn_gemm2_wmma(const float* __restrict__ h,
                               const float* __restrict__ W2,
                               float* __restrict__ hw) {
    __shared__ float sW[HIDDEN * NUM_CLASSES];    // 2 KB
    const int tid = threadIdx.x;
    for (int i = tid; i < HIDDEN * NUM_CLASSES; i += blockDim.x)
        sW[i] = W2[i];
    __syncthreads();

    const int wave = blockIdx.x * (blockDim.x >> 5) + (tid >> 5);
    const int lane = tid & 31;
    if (wave * 16 >= N_NODES) return;

    const int half = lane >> 4;
    const int m    = lane & 15;
    const float* hrow = h + (size_t)(wave * 16 + m) * HIDDEN + 2 * half;

    v8f c0 = {0.f, 0.f, 0.f, 0.f, 0.f, 0.f, 0.f, 0.f};
    v8f c1 = {0.f, 0.f, 0.f, 0.f, 0.f, 0.f, 0.f, 0.f};
#pragma unroll
    for (int k0 = 0; k0 < HIDDEN; k0 += 4) {
        v2f a = *(const v2f*)(hrow + k0);
        const int kr = k0 + 2 * half;
        v2f b0, b1;
        b0.x = sW[kr * NUM_CLASSES + m];
        b0.y = sW[(kr + 1) * NUM_CLASSES + m];
        b1.x = sW[kr * NUM_CLASSES + 16 + m];
        b1.y = sW[(kr + 1) * NUM_CLASSES + 16 + m];
        c0 = __builtin_amdgcn_wmma_f32_16x16x4_f32(
                false, a, false, b0, (short)0, c0, false, false);
        c1 = __builtin_amdgcn_wmma_f32_16x16x4_f32(
                false, a, false, b1, (short)0, c1, false, false);
    }

#pragma unroll
    for (int r = 0; r < 8; ++r) {
        const int rowD = wave * 16 + r + 8 * half;
        hw[(size_t)rowD * NUM_CLASSES + m]      = c0[r];
        hw[(size_t)rowD * NUM_CLASSES + 16 + m] = c1[r];
    }
}

// ---------------------------------------------------------------------------
// Scatter layer 2: out[dst] += hw[src] * norm   (32 features)
// ---------------------------------------------------------------------------
__global__ void gcn_scatter2_kernel(const long long* __restrict__ ei,
                                    const float* __restrict__ dinv,
                                    const float* __restrict__ hw,
                                    float* __restrict__ out) {
    int t = blockIdx.x * blockDim.x + threadIdx.x;
    if (t >= E_TOT) return;
    int s, d;
    if (t < N_EDGES) { s = (int)ei[t]; d = (int)ei[N_EDGES + t]; }
    else             { s = t - N_EDGES; d = s; }
    const float nrm = dinv[s] * dinv[d];
    const float4* xs = (const float4*)(hw + (size_t)s * NUM_CLASSES);
    float* ad = out + (size_t)d * NUM_CLASSES;
#pragma unroll
    for (int q = 0; q < 8; ++q) {
        float4 v = xs[q];
        atomicAdd(ad + 4 * q + 0, v.x * nrm);
        atomicAdd(ad + 4 * q + 1, v.y * nrm);
        atomicAdd(ad + 4 * q + 2, v.z * nrm);
        atomicAdd(ad + 4 * q + 3, v.w * nrm);
    }
}

// ---------------------------------------------------------------------------
extern "C" void kernel_launch(void* const* d_in, const int* in_sizes, int n_in,
                              void* d_out, int out_size, void* d_ws, size_t ws_size,
                              hipStream_t stream) {
    (void)in_sizes; (void)n_in; (void)out_size; (void)ws_size;

    const float*     x  = (const float*)d_in[0];          // [N, 512]
    const long long* ei = (const long long*)d_in[1];      // [2, E] int64
    const float*     W1 = (const float*)d_in[2];          // [512, 16]
    const float*     b1 = (const float*)d_in[3];          // [16]
    const float*     W2 = (const float*)d_in[4];          // [16, 32]
    const float*     b2 = (const float*)d_in[5];          // [32]

    float* out = (float*)d_out;                               // [N, 32]
    float* h   = out + (size_t)N_NODES * NUM_CLASSES;         // [N, 16]

    // Workspace layout (floats): dinv | xw | agg1 | hw  -> 65*N floats = 26 MB
    float* dinv = (float*)d_ws;
    float* xw   = dinv + N_NODES;
    float* agg1 = xw   + (size_t)N_NODES * HIDDEN;
    float* hw   = agg1 + (size_t)N_NODES * HIDDEN;

    const int B = 256;
    const int gInit    = (N_NODES * NUM_CLASSES + B - 1) / B;
    const int gEdges   = (E_TOT + B - 1) / B;
    const int gNodes   = (N_NODES + B - 1) / B;
    const int gH       = (N_NODES * HIDDEN + B - 1) / B;
    const int tiles    = N_NODES / 16;                    // 6250 exactly
    const int gGemm    = (tiles + (B / 32) - 1) / (B / 32);

    gcn_init_kernel<<<gInit, B, 0, stream>>>(dinv, agg1, out, b1, b2);
    gcn_degree_kernel<<<gEdges, B, 0, stream>>>(ei, dinv);
    gcn_dinv_kernel<<<gNodes, B, 0, stream>>>(dinv);
    gcn_gemm1_wmma<<<gGemm, B, 0, stream>>>(x, W1, xw);
    gcn_scatter1_kernel<<<gEdges, B, 0, stream>>>(ei, dinv, xw, agg1);
    gcn_relu_kernel<<<gH, B, 0, stream>>>(agg1, h);
    gcn_gemm2_wmma<<<gGemm, B, 0, stream>>>(h, W2, hw);
    gcn_scatter2_kernel<<<gEdges, B, 0, stream>>>(ei, dinv, hw, out);
}